// CrissCrossAttention_32736240730147
// MI455X (gfx1250) — compile-verified
//
#include <hip/hip_runtime.h>
#include <hip/hip_bf16.h>

#define S_LEN 512
#define B_N   8
#define D_DIM 64
#define FEAT  32
#define SDOWN 128   // S / DSR
#define UP_N  12
#define CF    3
#define PI_F  3.14159265358979323846f

typedef float v2f __attribute__((ext_vector_type(2)));
typedef float v8f __attribute__((ext_vector_type(8)));

// ---------------- deterministic block reductions (256 threads) ----------------
__device__ __forceinline__ float blockReduceSum(float v, float* red) {
    int tid = threadIdx.x;
    red[tid] = v; __syncthreads();
    for (int off = 128; off > 0; off >>= 1) {
        if (tid < off) red[tid] += red[tid + off];
        __syncthreads();
    }
    float r = red[0]; __syncthreads();
    return r;
}
__device__ __forceinline__ float blockReduceMax(float v, float* red) {
    int tid = threadIdx.x;
    red[tid] = v; __syncthreads();
    for (int off = 128; off > 0; off >>= 1) {
        if (tid < off) red[tid] = fmaxf(red[tid], red[tid + off]);
        __syncthreads();
    }
    float r = red[0]; __syncthreads();
    return r;
}
__device__ __forceinline__ float blockReduceMin(float v, float* red) {
    int tid = threadIdx.x;
    red[tid] = v; __syncthreads();
    for (int off = 128; off > 0; off >>= 1) {
        if (tid < off) red[tid] = fminf(red[tid], red[tid + off]);
        __syncthreads();
    }
    float r = red[0]; __syncthreads();
    return r;
}

// ---------------- Kernel 1: norms + 3-bin DFT + 12-bin inverse synthesis ------
// one block per (b, f); 256 threads
__global__ void cca_prep_kernel(const float* __restrict__ x,
                                const float* __restrict__ fw_r, const float* __restrict__ fw_i,
                                const float* __restrict__ fb_r, const float* __restrict__ fb_i,
                                float* __restrict__ img)
{
    const int bf = blockIdx.x;
    const int b  = bf >> 5;        // / FEAT
    const int f  = bf & 31;        // % FEAT
    const int tid = threadIdx.x;

    __shared__ float xs[S_LEN];
    __shared__ float red[256];
    __shared__ float upsh[2 * UP_N];   // 12 real, 12 imag

    for (int s = tid; s < S_LEN; s += 256)
        xs[s] = x[(s * B_N + b) * D_DIM + f];
    __syncthreads();

    // full-rate stats over S (ddof=1)
    float a1 = 0.f, a2 = 0.f;
    for (int s = tid; s < S_LEN; s += 256) { float u = xs[s]; a1 += u; a2 += u * u; }
    a1 = blockReduceSum(a1, red);
    a2 = blockReduceSum(a2, red);
    const float xm  = a1 / (float)S_LEN;
    const float xv  = (a2 - (float)S_LEN * xm * xm) / (float)(S_LEN - 1) + 1e-5f;
    const float xinv = 1.0f / sqrtf(xv);

    // downsampled stats over S/4 (ddof=1)
    float d1 = 0.f, d2 = 0.f;
    for (int t = tid; t < SDOWN; t += 256) { float u = xs[4 * t]; d1 += u; d2 += u * u; }
    d1 = blockReduceSum(d1, red);
    d2 = blockReduceSum(d2, red);
    const float fm   = d1 / (float)SDOWN;
    const float fv   = (d2 - (float)SDOWN * fm * fm) / (float)(SDOWN - 1) + 1e-5f;
    const float fsd  = sqrtf(fv);
    const float finv = 1.0f / fsd;

    // DFT bins c = 0,1,2 of fn[t] (rfft convention: sum fn * e^{-2*pi*i*c*t/128})
    float sr0 = 0.f, sr1 = 0.f, sr2 = 0.f, si1 = 0.f, si2 = 0.f;
    for (int t = tid; t < SDOWN; t += 256) {
        float fn = (xs[4 * t] - fm) * finv;
        sr0 += fn;
        float ang = (2.0f * PI_F / 128.0f) * (float)t;
        float sn, cs;
        __sincosf(ang, &sn, &cs);
        sr1 += fn * cs;          si1 -= fn * sn;
        float cs2 = 2.0f * cs * cs - 1.0f;   // cos(2a)
        float sn2 = 2.0f * sn * cs;          // sin(2a)
        sr2 += fn * cs2;         si2 -= fn * sn2;
    }
    sr0 = blockReduceSum(sr0, red);
    sr1 = blockReduceSum(sr1, red);
    si1 = blockReduceSum(si1, red);
    sr2 = blockReduceSum(sr2, red);
    si2 = blockReduceSum(si2, red);

    // up[k] = sum_c Wc[k,c]*spec[c] + bc[k]   (complex)
    if (tid == 0) {
        float Sr[CF] = { sr0, sr1, sr2 };
        float Si[CF] = { 0.f, si1, si2 };
        for (int kk = 0; kk < UP_N; ++kk) {
            float ur = fb_r[kk], ui = fb_i[kk];
            for (int c = 0; c < CF; ++c) {
                float wr = fw_r[kk * CF + c];
                float wi = fw_i[kk * CF + c];
                ur += wr * Sr[c] - wi * Si[c];
                ui += wr * Si[c] + wi * Sr[c];
            }
            upsh[kk]        = ur;
            upsh[UP_N + kk] = ui;
        }
    }
    __syncthreads();

    // irfft with only bins 0..11 nonzero (Im of bin 0 dropped), *DSR, then de-normalize
    const float scale = 4.0f / (float)S_LEN;   // DSR / S
    for (int s = tid; s < S_LEN; s += 256) {
        float acc = upsh[0];
        #pragma unroll
        for (int kk = 1; kk < UP_N; ++kk) {
            float ang = (2.0f * PI_F / 512.0f) * (float)((kk * s) & 511);
            float sn, cs;
            __sincosf(ang, &sn, &cs);
            acc += 2.0f * (upsh[kk] * cs - upsh[UP_N + kk] * sn);
        }
        float xy = acc * scale * fsd + fm;
        float xn = (xs[s] - xm) * xinv;
        float* row = img + (b * S_LEN + s) * D_DIM;
        row[2 * f]     = xn;
        row[2 * f + 1] = xy;
    }
}

// ---------------- Kernel 2: Q/K/V projections via V_WMMA_F32_16X16X4_F32 ------
// one wave per 16x16 output tile; grid = (M/16, N/16, 3)
__global__ void cca_qkv_wmma_kernel(const float* __restrict__ img,
                                    const float* __restrict__ Wq,
                                    const float* __restrict__ Wk,
                                    const float* __restrict__ Wv,
                                    float* __restrict__ q,
                                    float* __restrict__ k,
                                    float* __restrict__ v)
{
    const int lane  = threadIdx.x;          // 0..31
    const int m0    = blockIdx.x * 16;      // row tile in (B*S) = 4096
    const int n0    = blockIdx.y * 16;      // col tile in 64
    const int which = blockIdx.z;

    const float* W   = (which == 0) ? Wq : (which == 1) ? Wk : Wv;
    float*       out = (which == 0) ? q  : (which == 1) ? k  : v;

    const int half  = lane >> 4;
    const int idx16 = lane & 15;

    // A fragment rows: img[(m0+idx16), :]   (M = idx16 for both lane halves)
    const float* arow = img + (m0 + idx16) * D_DIM;
    // B = W^T : B[kk][n] = W[n][kk]; lane's n = idx16
    const float* brow = W + (n0 + idx16) * D_DIM;

    v8f acc = {};
    #pragma unroll
    for (int k0 = 0; k0 < D_DIM; k0 += 4) {
        const int kk = k0 + 2 * half;        // lanes 0-15: K = k0,k0+1 ; lanes 16-31: K = k0+2,k0+3
        v2f afrag, bfrag;
        afrag[0] = arow[kk];
        afrag[1] = arow[kk + 1];
        bfrag[0] = brow[kk];
        bfrag[1] = brow[kk + 1];
        acc = __builtin_amdgcn_wmma_f32_16x16x4_f32(
                  /*neg_a=*/false, afrag,
                  /*neg_b=*/false, bfrag,
                  /*c_mod=*/(short)0, acc,
                  /*reuse_a=*/false, /*reuse_b=*/false);
    }

    // C/D layout: VGPR r, lanes 0-15 -> M = r, lanes 16-31 -> M = r+8 ; N = lane&15
    #pragma unroll
    for (int r = 0; r < 8; ++r) {
        const int m = m0 + r + 8 * half;
        out[m * D_DIM + n0 + idx16] = acc[r];
    }
}

// ---------------- Kernel 3: criss-cross joint softmax + aggregation ----------
// one block per (b, w); 256 threads, each handles 2 h values
__global__ void cca_attn_kernel(const float* __restrict__ q,
                                const float* __restrict__ k,
                                const float* __restrict__ v,
                                const float* __restrict__ gamma,
                                float* __restrict__ out)
{
    const int b   = blockIdx.x >> 6;   // / 64
    const int w   = blockIdx.x & 63;   // % 64
    const int tid = threadIdx.x;

    __shared__ float kcol[S_LEN];
    __shared__ float vcol[S_LEN];
    __shared__ float red[256];

    const float* kb = k + b * S_LEN * D_DIM;
    const float* vb = v + b * S_LEN * D_DIM;

    for (int g = tid; g < S_LEN; g += 256) {
        kcol[g] = kb[g * D_DIM + w];
        vcol[g] = vb[g * D_DIM + w];
    }
    __syncthreads();

    float cmx = -3.402823466e+38f, cmn = 3.402823466e+38f;
    for (int g = tid; g < S_LEN; g += 256) {
        cmx = fmaxf(cmx, kcol[g]);
        cmn = fminf(cmn, kcol[g]);
    }
    cmx = blockReduceMax(cmx, red);
    cmn = blockReduceMin(cmn, red);

    const float g0 = gamma[0];

    for (int h = tid; h < S_LEN; h += 256) {
        const float* krow = kb + h * D_DIM;
        const float* vrow = vb + h * D_DIM;
        const float  qp   = q[(b * S_LEN + h) * D_DIM + w];

        // joint max of logits: q * (q>=0 ? max(k) : min(k)) over the 576-set
        float rmx = cmx, rmn = cmn;
        for (int j = 0; j < D_DIM; ++j) {
            float kv = krow[j];
            rmx = fmaxf(rmx, kv);
            rmn = fminf(rmn, kv);
        }
        const float m = (qp >= 0.f) ? qp * rmx : qp * rmn;

        float den = 0.f, num = 0.f;
        for (int g = 0; g < S_LEN; ++g) {
            float e = __expf(qp * kcol[g] - m);
            den += e;
            num += e * vcol[g];
        }
        for (int j = 0; j < D_DIM; ++j) {
            float e = __expf(qp * krow[j] - m);
            den += e;
            num += e * vrow[j];
        }
        // output transposed to (S, B, D)
        out[(h * B_N + b) * D_DIM + w] = g0 * num / den;
    }
}

// ------------------------------------------------------------------------------
extern "C" void kernel_launch(void* const* d_in, const int* in_sizes, int n_in,
                              void* d_out, int out_size, void* d_ws, size_t ws_size,
                              hipStream_t stream)
{
    const float* x     = (const float*)d_in[0];
    const float* Wq    = (const float*)d_in[1];
    const float* Wk    = (const float*)d_in[2];
    const float* Wv    = (const float*)d_in[3];
    const float* fw_r  = (const float*)d_in[4];
    const float* fw_i  = (const float*)d_in[5];
    const float* fb_r  = (const float*)d_in[6];
    const float* fb_i  = (const float*)d_in[7];
    const float* gamma = (const float*)d_in[8];
    float*       out   = (float*)d_out;

    const size_t N = (size_t)B_N * S_LEN * D_DIM;   // 262144
    float* img = (float*)d_ws;
    float* q   = img + N;
    float* k   = q + N;
    float* v   = k + N;

    // 1) spectral prep -> img (B, S, 64)
    cca_prep_kernel<<<B_N * FEAT, 256, 0, stream>>>(x, fw_r, fw_i, fb_r, fb_i, img);

    // 2) q/k/v = img @ W^T via f32 WMMA (M=4096, N=64, K=64)
    dim3 g2((B_N * S_LEN) / 16, D_DIM / 16, 3);
    cca_qkv_wmma_kernel<<<g2, 32, 0, stream>>>(img, Wq, Wk, Wv, q, k, v);

    // 3) criss-cross attention
    cca_attn_kernel<<<B_N * D_DIM, 256, 0, stream>>>(q, k, v, gamma, out);
}